// GCN_26903675142661
// MI455X (gfx1250) — compile-verified
//
#include <hip/hip_runtime.h>
#include <hip/hip_bf16.h>

#define N_NODES 100000
#define N_EDGES 1000000
#define N_GRAPHS 1000
#define EMBED 64
#define OUT_DIM 128
#define N_CONV 3
#define BN_EPS 1e-5f

typedef __attribute__((ext_vector_type(2))) float v2f;
typedef __attribute__((ext_vector_type(8))) float v8f;

// ---------------- utility ----------------
__global__ void zero_kernel(float* __restrict__ p, int n) {
    for (int i = blockIdx.x * blockDim.x + threadIdx.x; i < n;
         i += gridDim.x * blockDim.x)
        p[i] = 0.0f;
}

// h[n][c] = atom_embed[nfeat[n]][c]
__global__ void embed_kernel(const int* __restrict__ nfeat,
                             const float* __restrict__ atom_embed,
                             float* __restrict__ h) {
    int idx = blockIdx.x * blockDim.x + threadIdx.x;
    if (idx >= N_NODES * EMBED) return;
    int n = idx >> 6, c = idx & 63;
    h[idx] = atom_embed[nfeat[n] * EMBED + c];
}

// deg[d] = #edges with dst == d (dst is layer-invariant: compute once)
__global__ void deg_kernel(const int* __restrict__ dst, float* __restrict__ deg) {
    int e = blockIdx.x * blockDim.x + threadIdx.x;
    if (e < N_EDGES) atomicAdd(&deg[dst[e]], 1.0f);
}

// wave-per-edge: msg = relu(bond[efeat[e]] + h[src[e]]); agg[dst[e]] += msg
// lane l handles channels 2l, 2l+1 -> contiguous 256B per wave (L2-resident)
__global__ __launch_bounds__(256) void edge_kernel(
    const int* __restrict__ efeat, const int* __restrict__ src,
    const int* __restrict__ dst, const float* __restrict__ bond,
    const float* __restrict__ h, float* __restrict__ agg) {
    int e = blockIdx.x * 8 + (threadIdx.x >> 5);
    if (e >= N_EDGES) return;
    int lane = threadIdx.x & 31;
    int s  = src[e];
    int d  = dst[e];
    int bf = efeat[e];
    int c = lane * 2;
    const float2 be = *(const float2*)(bond + bf * EMBED + c);
    const float2 hv = *(const float2*)(h + (size_t)s * EMBED + c);
    float m0 = be.x + hv.x;
    float m1 = be.y + hv.y;
    m0 = m0 > 0.0f ? m0 : 0.0f;
    m1 = m1 > 0.0f ? m1 : 0.0f;
    float* ap = agg + (size_t)d * EMBED + c;
    atomicAdd(ap,     m0);
    atomicAdd(ap + 1, m1);
}

// y = (agg/max(deg,1) + h) @ W + b, written in-place over agg.
// Fused BN partial stats: bnsum[c] += sum_y, bnsumsq[c] += sum_y2.
// One block = 16 node rows; 4 waves, wave w computes output cols [16w,16w+16)
// via 16x V_WMMA_F32_16X16X4_F32 (K=64).
__global__ __launch_bounds__(128) void gemm_bn_kernel(
    const float* __restrict__ h, float* __restrict__ agg,
    const float* __restrict__ deg, const float* __restrict__ W,
    const float* __restrict__ bvec, float* __restrict__ bnsum,
    float* __restrict__ bnsumsq) {
    __shared__ float As[16][EMBED + 1];  // stride 65 -> conflict-free
    const int tileRow = blockIdx.x * 16;
    const int tid = threadIdx.x;
    for (int i = tid; i < 16 * EMBED; i += 128) {
        int r = i >> 6, c = i & 63;
        int n = tileRow + r;
        float dm = deg[n];
        dm = dm > 1.0f ? dm : 1.0f;
        As[r][c] = agg[(size_t)n * EMBED + c] / dm + h[(size_t)n * EMBED + c];
    }
    __syncthreads();
    const int wave = tid >> 5;
    const int lane = tid & 31;
    const int colBase = wave * 16;
    const int half = lane >> 4;   // 0: K pair {0,1}, 1: K pair {2,3}
    const int l16  = lane & 15;
    v8f acc = {};
    for (int kk = 0; kk < 16; ++kk) {
        int k0 = kk * 4 + half * 2;
        v2f a, b;
        a.x = As[l16][k0];
        a.y = As[l16][k0 + 1];
        b.x = W[k0 * EMBED + colBase + l16];
        b.y = W[(k0 + 1) * EMBED + colBase + l16];
        acc = __builtin_amdgcn_wmma_f32_16x16x4_f32(
            false, a, false, b, (short)0, acc, false, false);
    }
    const int col = colBase + l16;
    const float bias = bvec[col];
    float s = 0.0f, s2 = 0.0f;
    for (int v = 0; v < 8; ++v) {
        int r = v + half * 8;
        float y = acc[v] + bias;
        agg[(size_t)(tileRow + r) * EMBED + col] = y;
        s += y;
        s2 += y * y;
    }
    atomicAdd(&bnsum[col], s);
    atomicAdd(&bnsumsq[col], s2);
}

// h = h + maybe_relu( (y - mu) * rsqrt(var+eps) * gamma + beta ), in-place on h
__global__ void bn_apply_kernel(const float* __restrict__ y,
                                float* __restrict__ h,
                                const float* __restrict__ bnsum,
                                const float* __restrict__ bnsumsq,
                                const float* __restrict__ gamma,
                                const float* __restrict__ beta,
                                int do_relu) {
    int idx = blockIdx.x * blockDim.x + threadIdx.x;
    if (idx >= N_NODES * EMBED) return;
    int c = idx & 63;
    const float invN = 1.0f / (float)N_NODES;
    float mu  = bnsum[c] * invN;
    float var = bnsumsq[c] * invN - mu * mu;
    float rs = rsqrtf(var + BN_EPS);
    float v = (y[idx] - mu) * rs * gamma[c] + beta[c];
    if (do_relu) v = v > 0.0f ? v : 0.0f;
    h[idx] = h[idx] + v;
}

// gsum[gid[n]][c] += h[n][c]; gcnt[gid[n]] += 1
__global__ void pool_kernel(const float* __restrict__ h,
                            const int* __restrict__ gid,
                            float* __restrict__ gsum, float* __restrict__ gcnt) {
    int idx = blockIdx.x * blockDim.x + threadIdx.x;
    if (idx >= N_NODES * EMBED) return;
    int n = idx >> 6, c = idx & 63;
    int g = gid[n];
    atomicAdd(&gsum[g * EMBED + c], h[idx]);
    if (c == 0) atomicAdd(&gcnt[g], 1.0f);
}

// out[g][o] = (gsum[g]/max(gcnt,1)) @ pred_W + pred_b   [1000,64]x[64,128]
// one block = 16 graphs, 8 waves cover the 128 output columns.
__global__ __launch_bounds__(256) void final_gemm_kernel(
    const float* __restrict__ gsum, const float* __restrict__ gcnt,
    const float* __restrict__ predW, const float* __restrict__ predb,
    float* __restrict__ out) {
    __shared__ float As[16][EMBED + 1];
    const int tileRow = blockIdx.x * 16;
    const int tid = threadIdx.x;
    for (int i = tid; i < 16 * EMBED; i += 256) {
        int r = i >> 6, c = i & 63;
        int g = tileRow + r;
        float v = 0.0f;
        if (g < N_GRAPHS) {
            float cnt = gcnt[g];
            cnt = cnt > 1.0f ? cnt : 1.0f;
            v = gsum[g * EMBED + c] / cnt;
        }
        As[r][c] = v;
    }
    __syncthreads();
    const int wave = tid >> 5;
    const int lane = tid & 31;
    const int colBase = wave * 16;
    const int half = lane >> 4;
    const int l16  = lane & 15;
    v8f acc = {};
    for (int kk = 0; kk < 16; ++kk) {
        int k0 = kk * 4 + half * 2;
        v2f a, b;
        a.x = As[l16][k0];
        a.y = As[l16][k0 + 1];
        b.x = predW[k0 * OUT_DIM + colBase + l16];
        b.y = predW[(k0 + 1) * OUT_DIM + colBase + l16];
        acc = __builtin_amdgcn_wmma_f32_16x16x4_f32(
            false, a, false, b, (short)0, acc, false, false);
    }
    const int col = colBase + l16;
    const float bias = predb[col];
    for (int v = 0; v < 8; ++v) {
        int r = tileRow + v + half * 8;
        if (r < N_GRAPHS) out[r * OUT_DIM + col] = acc[v] + bias;
    }
}

extern "C" void kernel_launch(void* const* d_in, const int* in_sizes, int n_in,
                              void* d_out, int out_size, void* d_ws, size_t ws_size,
                              hipStream_t stream) {
    (void)in_sizes; (void)n_in; (void)out_size; (void)ws_size;
    const int*   nfeat      = (const int*)d_in[0];
    const int*   efeat      = (const int*)d_in[1];
    const int*   src        = (const int*)d_in[2];
    const int*   dst        = (const int*)d_in[3];
    const int*   graph_ids  = (const int*)d_in[4];
    const float* atom_embed = (const float*)d_in[5];
    const float* bond_embed = (const float*)d_in[6];   // [3,5,64]
    const float* conv_W     = (const float*)d_in[7];   // [3,64,64]
    const float* conv_b     = (const float*)d_in[8];   // [3,64]
    const float* bn_gamma   = (const float*)d_in[9];   // [3,64]
    const float* bn_beta    = (const float*)d_in[10];  // [3,64]
    const float* pred_W     = (const float*)d_in[11];  // [64,128]
    const float* pred_b     = (const float*)d_in[12];  // [128]
    float* out = (float*)d_out;

    // workspace carve (floats)
    float* h    = (float*)d_ws;                 // N*64
    float* agg  = h    + (size_t)N_NODES * EMBED;
    float* deg  = agg  + (size_t)N_NODES * EMBED;      // N
    float* bnst = deg  + N_NODES;                      // 128 (sum|sumsq)
    float* gsum = bnst + 128;                          // G*64
    float* gcnt = gsum + (size_t)N_GRAPHS * EMBED;     // G

    const int NE = N_NODES * EMBED;   // 6,400,000
    dim3 b256(256);

    // degrees (once) + atom embedding
    zero_kernel<<<1024, b256, 0, stream>>>(deg, N_NODES);
    embed_kernel<<<(NE + 255) / 256, b256, 0, stream>>>(nfeat, atom_embed, h);
    deg_kernel<<<(N_EDGES + 255) / 256, b256, 0, stream>>>(dst, deg);

    for (int i = 0; i < N_CONV; ++i) {
        zero_kernel<<<2048, b256, 0, stream>>>(agg, NE);
        zero_kernel<<<1, dim3(128), 0, stream>>>(bnst, 128);
        edge_kernel<<<(N_EDGES + 7) / 8, b256, 0, stream>>>(
            efeat, src, dst, bond_embed + i * 5 * EMBED, h, agg);
        gemm_bn_kernel<<<N_NODES / 16, dim3(128), 0, stream>>>(
            h, agg, deg, conv_W + i * EMBED * EMBED, conv_b + i * EMBED,
            bnst, bnst + 64);
        bn_apply_kernel<<<(NE + 255) / 256, b256, 0, stream>>>(
            agg, h, bnst, bnst + 64, bn_gamma + i * EMBED, bn_beta + i * EMBED,
            (i != N_CONV - 1) ? 1 : 0);
    }

    // graph mean pooling + prediction head
    zero_kernel<<<256, b256, 0, stream>>>(gsum, N_GRAPHS * EMBED + N_GRAPHS);
    pool_kernel<<<(NE + 255) / 256, b256, 0, stream>>>(h, graph_ids, gsum, gcnt);
    final_gemm_kernel<<<(N_GRAPHS + 15) / 16, b256, 0, stream>>>(
        gsum, gcnt, pred_W, pred_b, out);
}